// fpm_85907935855332
// MI455X (gfx1250) — compile-verified
//
#include <hip/hip_runtime.h>
#include <hip/hip_bf16.h>

// ---------------------------------------------------------------------------
// FPM gradient step for MI455X (gfx1250, wave32).
//  H=W=512, NLEDS=96, NMEAS=24.
//  Pipeline:
//   P1 rowFFT(pw*field)            -> tmpA
//   P2 colFFT * P * col-iFFT       -> outb     (2D mask fused into one pass)
//   P3 row-iFFT (in place)         -> outb = "out"
//   P4 WMMA fused einsums:  y=C|out|^2, mM=|C meas|, Ajx, Wl=C^T Ajx -> Wl
//   P5 rowFFT(conj(pw)*out*Wl)     -> tmpA
//   P6 colFFT * pupils * col-iFFT  -> tmpA (in place)
//   P7 row-iFFT, sum over 96 LEDs in registers, field - ALPHA*g -> d_out
//  FFTs: radix-8 Stockham (512 = 8^3), 64 lanes per transform, LDS resident,
//  with pad-per-8 index map to kill the 16-way stage-0 scatter conflicts.
//  Async global->LDS copies (ASYNCcnt) used where loads are pure copies.
// ---------------------------------------------------------------------------

#define HH 512
#define WW 512
#define NLEDS 96
#define NMEAS 24
#define HWPIX (512*512)
#define ALPHA 0.1f
#define EPSV 1e-7f

typedef __attribute__((ext_vector_type(2))) float v2f;
typedef __attribute__((ext_vector_type(8))) float v8f;

// ---- async global->LDS (CDNA5 GLOBAL_LOAD_ASYNC_TO_LDS_B64, ASYNCcnt) ----
#if defined(__has_builtin)
#if __has_builtin(__builtin_amdgcn_global_load_async_to_lds_b64)
#define HAVE_ASYNC_LDS 1
#endif
#endif

#ifdef HAVE_ASYNC_LDS
// Exact param types per hipcc diagnostic:
//   arg0: int __vector(2) __device__* (AS1 global), arg1: AS3 (LDS)
typedef int v2i_gcc __attribute__((vector_size(8)));
typedef __attribute__((address_space(1))) v2i_gcc as1_v2i;
typedef __attribute__((address_space(3))) v2i_gcc as3_v2i;
__device__ __forceinline__ void async_cp_b64(float2* lds, const float2* g) {
    __builtin_amdgcn_global_load_async_to_lds_b64(
        (as1_v2i*)g, (as3_v2i*)lds, 0, 0);
}
__device__ __forceinline__ void async_wait_all() {
#if __has_builtin(__builtin_amdgcn_s_wait_asynccnt)
    __builtin_amdgcn_s_wait_asynccnt(0);
#else
    asm volatile("s_wait_asynccnt 0" ::: "memory");
#endif
}
#else
__device__ __forceinline__ void async_cp_b64(float2* lds, const float2* g) { *lds = *g; }
__device__ __forceinline__ void async_wait_all() {}
#endif

__device__ __forceinline__ float2 cmul(float2 a, float2 b) {
    return make_float2(a.x*b.x - a.y*b.y, a.x*b.y + a.y*b.x);
}
__device__ __forceinline__ float2 cadd(float2 a, float2 b) { return make_float2(a.x+b.x, a.y+b.y); }
__device__ __forceinline__ float2 csub(float2 a, float2 b) { return make_float2(a.x-b.x, a.y-b.y); }
// multiply by (S*i): S=-1 -> *( -i ), S=+1 -> *( +i )
template<int S>
__device__ __forceinline__ float2 muli(float2 a) {
    return (S > 0) ? make_float2(-a.y, a.x) : make_float2(a.y, -a.x);
}

// Padded LDS index: one float2 gap every 8 elements -> <=2-way bank conflicts
// in every Stockham phase (unpadded stage-0 scatter is 16-way).
__device__ __forceinline__ int pidx(int i) { return i + (i >> 3); }
#define ROWBUF 576   // 512 + 64 pad slots

// 8-point DFT, sign S (-1 forward, +1 inverse), natural-order in/out (DIF).
template<int S>
__device__ __forceinline__ void dft8(float2 a[8]) {
    const float c = 0.70710678118654752f;
    const float sc = (S > 0) ? c : -c;
    float2 u0 = cadd(a[0], a[4]), u1 = cadd(a[1], a[5]);
    float2 u2 = cadd(a[2], a[6]), u3 = cadd(a[3], a[7]);
    float2 v0 = csub(a[0], a[4]);
    float2 v1 = cmul(csub(a[1], a[5]), make_float2(c, sc));
    float2 v2 = muli<S>(csub(a[2], a[6]));
    float2 v3 = cmul(csub(a[3], a[7]), make_float2(-c, sc));
    float2 p0 = cadd(u0, u2), p1 = cadd(u1, u3);
    float2 q0 = csub(u0, u2), q1 = muli<S>(csub(u1, u3));
    a[0] = cadd(p0, p1); a[4] = csub(p0, p1);
    a[2] = cadd(q0, q1); a[6] = csub(q0, q1);
    float2 r0 = cadd(v0, v2), r1 = cadd(v1, v3);
    float2 t0 = csub(v0, v2), t1 = muli<S>(csub(v1, v3));
    a[1] = cadd(r0, r1); a[5] = csub(r0, r1);
    a[3] = cadd(t0, t1); a[7] = csub(t0, t1);
}

// In-place 512-point Stockham FFT on LDS. PAD=true uses the padded index map
// (element stride 1); PAD=false uses raw element stride `st` (column pitch).
// 64 lanes per transform (ft = 0..63). Every thread of the block must call
// this (block-wide barriers separate read / write phases of each stage).
template<int S, bool PAD>
__device__ void fft512(float2* buf, int st, int ft) {
    float2 a[8];
#pragma unroll
    for (int s = 0; s < 3; ++s) {
        const int lsLog = 3 * s;         // Ls = 1, 8, 64
        const int Ls = 1 << lsLog;
        const int l = ft & (Ls - 1);
        const int b = ft >> lsLog;
#pragma unroll
        for (int r = 0; r < 8; ++r) {
            const int gi = ft + r * 64;
            a[r] = buf[PAD ? pidx(gi) : gi * st];
        }
        if (s > 0) {
            float base = (float)S * 6.28318530717958647f * (float)l * (1.0f / (float)(8 * Ls));
#pragma unroll
            for (int r = 1; r < 8; ++r) {
                float sn, cn;
                __sincosf(base * (float)r, &sn, &cn);
                a[r] = cmul(a[r], make_float2(cn, sn));
            }
        }
        dft8<S>(a);
        __syncthreads();                  // all reads done
#pragma unroll
        for (int r = 0; r < 8; ++r) {
            const int wi = b * Ls * 8 + r * Ls + l;
            buf[PAD ? pidx(wi) : wi * st] = a[r];
        }
        __syncthreads();                  // all writes visible
    }
}

// ---------------------------------------------------------------------------
// Row-pass kernel: 4 rows per 256-thread block, fused input formation.
//  mode 0: v = pw*field, forward FFT
//  mode 1: v = in,       inverse FFT (scaled 1/512)   [async LDS copy]
//  mode 2: v = conj(pw)*out*Wl, forward FFT
// ---------------------------------------------------------------------------
__global__ __launch_bounds__(256) void fpm_rowfft_kernel(
    const float2* pw, const float2* field, const float2* inbuf,
    const float* wl, float2* dst, int mode)
{
    __shared__ float2 rows[4][ROWBUF];
    const int tid = threadIdx.x;
    const int grp = tid >> 6;
    const int ft  = tid & 63;
    const size_t rowIdx = (size_t)blockIdx.x * 4 + grp;   // in [0, 96*512)
    const size_t base = rowIdx * 512;
    const int h = (int)(rowIdx & 511);
#pragma unroll
    for (int r = 0; r < 8; ++r) {
        const int w = ft + r * 64;
        if (mode == 0) {
            rows[grp][pidx(w)] = cmul(pw[base + w], field[(size_t)h * 512 + w]);
        } else if (mode == 1) {
            async_cp_b64(&rows[grp][pidx(w)], &inbuf[base + w]);
        } else {
            float2 p = pw[base + w]; p.y = -p.y;
            float2 o = inbuf[base + w];
            float  s = wl[base + w];
            float2 v = cmul(p, o);
            v.x *= s; v.y *= s;
            rows[grp][pidx(w)] = v;
        }
    }
    if (mode == 1) async_wait_all();      // uniform branch: mode is a kernel arg
    __syncthreads();
    if (mode == 1) fft512<1, true>(&rows[grp][0], 1, ft);
    else           fft512<-1, true>(&rows[grp][0], 1, ft);
#pragma unroll
    for (int r = 0; r < 8; ++r) {
        const int w = ft + r * 64;
        float2 v = rows[grp][pidx(w)];
        if (mode == 1) { v.x *= (1.0f / 512.0f); v.y *= (1.0f / 512.0f); }
        dst[base + w] = v;
    }
}

// ---------------------------------------------------------------------------
// Column pass: colFFT -> mask multiply -> col-iFFT, fully LDS resident.
// 8 columns per block (pitch 9 breaks bank-conflict periodicity).
//  mode 0: mask = complex P[h,w] (broadcast over LEDs)
//  mode 1: mask = real pupils[l,h,w]
// ---------------------------------------------------------------------------
__global__ __launch_bounds__(256) void fpm_colpass_kernel(
    const float2* src, const float2* Pm, const float* pupils,
    float2* dst, int mode)
{
    __shared__ float2 tile[512 * 9];
    const int tid = threadIdx.x;
    const int l = blockIdx.y;
    const int colbase = blockIdx.x * 8;
    const size_t base = (size_t)l * HWPIX;
    for (int e = tid; e < 512 * 8; e += 256) {
        const int row = e >> 3, col = e & 7;
        async_cp_b64(&tile[row * 9 + col], &src[base + (size_t)row * 512 + colbase + col]);
    }
    async_wait_all();
    __syncthreads();
    const int grp = tid >> 6, ft = tid & 63;
    for (int cc = 0; cc < 2; ++cc)
        fft512<-1, false>(&tile[cc * 4 + grp], 9, ft);
    for (int e = tid; e < 512 * 8; e += 256) {
        const int row = e >> 3, col = e & 7;
        float2 v = tile[row * 9 + col];
        if (mode == 0) {
            v = cmul(v, Pm[(size_t)row * 512 + colbase + col]);
        } else {
            const float pu = pupils[base + (size_t)row * 512 + colbase + col];
            v.x *= pu; v.y *= pu;
        }
        tile[row * 9 + col] = v;
    }
    __syncthreads();
    for (int cc = 0; cc < 2; ++cc)
        fft512<1, false>(&tile[cc * 4 + grp], 9, ft);
    for (int e = tid; e < 512 * 8; e += 256) {
        const int row = e >> 3, col = e & 7;
        float2 v = tile[row * 9 + col];
        v.x *= (1.0f / 512.0f); v.y *= (1.0f / 512.0f);
        dst[base + (size_t)row * 512 + colbase + col] = v;
    }
}

// ---------------------------------------------------------------------------
// WMMA fused einsum kernel. Per wave: one 16-pixel tile.
//  y[24,16]  = C[24,96] @ |out|^2[96,16]        (V_WMMA_F32_16X16X4_F32)
//  mM[24,16] = |C[24,96] @ meas[96,16]|
//  Ajx = (sqrt(y+eps)-sqrt(mM+eps))/(sqrt(y+eps)+1e-10)
//  Wl[96,16] = C^T[96,24] @ Ajx[24,16]
// C is zero-padded to 32 rows in LDS and Ajx staged in a 32-row buffer so
// every fragment access is unconditional -> EXEC stays all-1s around WMMA
// (no saveexec/ds_load divergence in the inner loop).
// A 16x4 frag: lane(0..15)=M rows, vgpr{0,1}=K{0,1}; lanes 16..31: K{2,3}.
// B 4x16 frag: lane(0..15)=N cols, vgpr{0,1}=K{0,1}; lanes 16..31: K{2,3}.
// C/D 16x16:   vgpr v -> M = v + 8*(lane>=16); N = lane&15.
// ---------------------------------------------------------------------------
__global__ __launch_bounds__(256) void fpm_wmma_einsum_kernel(
    const float2* __restrict__ outb, const float* __restrict__ meas,
    const float* __restrict__ C, float* __restrict__ Wl)
{
    __shared__ float Cs[32 * NLEDS];             // rows 24..31 zero-padded
    __shared__ float AjxS[8][32 * 16];           // per-wave staging, padded rows
    const int tid = threadIdx.x;
    for (int e = tid; e < 32 * NLEDS; e += 256)
        Cs[e] = (e < NMEAS * NLEDS) ? C[e] : 0.0f;
    __syncthreads();

    const int wave = tid >> 5;
    const int lane = tid & 31;
    const int n  = lane & 15;
    const int hi = lane >> 4;                    // 0 or 1
    const size_t p0 = (size_t)blockIdx.x * 128 + (size_t)wave * 16;

    v8f accY0 = {}, accY1 = {}, accM0 = {}, accM1 = {};
    const int m  = lane & 15;
    const int m2 = 16 + m;                       // padded rows -> no branch
#pragma unroll
    for (int kk = 0; kk < 24; ++kk) {            // K = 96 in chunks of 4
        const int k0 = kk * 4 + 2 * hi;
        v2f a0, a1, bS, bM;
        a0.x = Cs[m * NLEDS + k0];
        a0.y = Cs[m * NLEDS + k0 + 1];
        a1.x = Cs[m2 * NLEDS + k0];
        a1.y = Cs[m2 * NLEDS + k0 + 1];
        const float2 o0 = outb[(size_t)k0 * HWPIX + p0 + n];
        const float2 o1 = outb[(size_t)(k0 + 1) * HWPIX + p0 + n];
        bS.x = o0.x * o0.x + o0.y * o0.y;
        bS.y = o1.x * o1.x + o1.y * o1.y;
        bM.x = meas[(size_t)k0 * HWPIX + p0 + n];
        bM.y = meas[(size_t)(k0 + 1) * HWPIX + p0 + n];
        accY0 = __builtin_amdgcn_wmma_f32_16x16x4_f32(false, a0, false, bS, (short)0, accY0, false, false);
        accM0 = __builtin_amdgcn_wmma_f32_16x16x4_f32(false, a0, false, bM, (short)0, accM0, false, false);
        accY1 = __builtin_amdgcn_wmma_f32_16x16x4_f32(false, a1, false, bS, (short)0, accY1, false, false);
        accM1 = __builtin_amdgcn_wmma_f32_16x16x4_f32(false, a1, false, bM, (short)0, accM1, false, false);
    }

    // Ajx in accumulator layout -> LDS (rows m, cols n), per-wave segment.
#pragma unroll
    for (int v = 0; v < 8; ++v) {
        const int mr = v + 8 * hi;               // rows 0..15
        float sy = __fsqrt_rn(accY0[v] + EPSV);
        float aj = (sy - __fsqrt_rn(fabsf(accM0[v]) + EPSV)) / (sy + 1e-10f);
        AjxS[wave][mr * 16 + n] = aj;
        float sy2 = __fsqrt_rn(accY1[v] + EPSV); // rows 16..31 (>=24 unused)
        float aj2 = (sy2 - __fsqrt_rn(fabsf(accM1[v]) + EPSV)) / (sy2 + 1e-10f);
        AjxS[wave][(16 + mr) * 16 + n] = aj2;
    }
    // order LDS writes before cross-lane reads (intra-wave, no s_barrier needed)
    asm volatile("s_wait_dscnt 0" ::: "memory");

    // Wl = C^T @ Ajx : 6 M-blocks of 16 LEDs, K = 24 in chunks of 4.
#pragma unroll
    for (int mb = 0; mb < 6; ++mb) {
        v8f acc = {};
        const int lrow = mb * 16 + m;            // LED index (A2 row)
#pragma unroll
        for (int kk = 0; kk < 6; ++kk) {
            const int k0 = kk * 4 + 2 * hi;      // measurement index (<24)
            v2f a, b;
            a.x = Cs[k0 * NLEDS + lrow];         // A2[l,m] = C[m,l]
            a.y = Cs[(k0 + 1) * NLEDS + lrow];
            b.x = AjxS[wave][k0 * 16 + n];
            b.y = AjxS[wave][(k0 + 1) * 16 + n];
            acc = __builtin_amdgcn_wmma_f32_16x16x4_f32(false, a, false, b, (short)0, acc, false, false);
        }
#pragma unroll
        for (int v = 0; v < 8; ++v) {
            const int lr = mb * 16 + v + 8 * hi;
            Wl[(size_t)lr * HWPIX + p0 + n] = acc[v];
        }
    }
}

// ---------------------------------------------------------------------------
// Final pass: row-iFFT of tmpA[l,h,:] for all 96 LEDs, accumulate g in
// registers, then new_field = field - ALPHA * g.
// ---------------------------------------------------------------------------
__global__ __launch_bounds__(256) void fpm_rowifft_accum_kernel(
    const float2* src, const float2* xin, float2* outp)
{
    __shared__ float2 rows[4][ROWBUF];
    const int tid = threadIdx.x;
    const int grp = tid >> 6;
    const int ft  = tid & 63;
    const int h = blockIdx.x * 4 + grp;
    float2 acc[8];
#pragma unroll
    for (int r = 0; r < 8; ++r) acc[r] = make_float2(0.0f, 0.0f);
    for (int l = 0; l < NLEDS; ++l) {
        const size_t base = ((size_t)l * 512 + h) * 512;
        if (l + 1 < NLEDS)
            __builtin_prefetch((const void*)(src + base + HWPIX + ft), 0, 1);
#pragma unroll
        for (int r = 0; r < 8; ++r)
            async_cp_b64(&rows[grp][pidx(ft + r * 64)], &src[base + ft + r * 64]);
        async_wait_all();
        __syncthreads();
        fft512<1, true>(&rows[grp][0], 1, ft);
#pragma unroll
        for (int r = 0; r < 8; ++r) {
            const float2 v = rows[grp][pidx(ft + r * 64)];
            acc[r].x += v.x * (1.0f / 512.0f);
            acc[r].y += v.y * (1.0f / 512.0f);
        }
        __syncthreads();                          // protect rows before next LED
    }
#pragma unroll
    for (int r = 0; r < 8; ++r) {
        const int w = ft + r * 64;
        const float2 xv = xin[(size_t)h * 512 + w];
        outp[(size_t)h * 512 + w] =
            make_float2(xv.x - ALPHA * acc[r].x, xv.y - ALPHA * acc[r].y);
    }
}

// ---------------------------------------------------------------------------
extern "C" void kernel_launch(void* const* d_in, const int* in_sizes, int n_in,
                              void* d_out, int out_size, void* d_ws, size_t ws_size,
                              hipStream_t stream) {
    const float2* x      = (const float2*)d_in[0];   // [512,512,2]
    const float2* pw     = (const float2*)d_in[1];   // [96,512,512,2]
    const float2* P      = (const float2*)d_in[2];   // [512,512,2]
    const float*  pupils = (const float*)d_in[3];    // [96,512,512]
    const float*  C      = (const float*)d_in[4];    // [24,96]
    const float*  meas   = (const float*)d_in[5];    // [96,512,512]

    char* ws = (char*)d_ws;
    const size_t volB = (size_t)NLEDS * HWPIX * sizeof(float2);  // 192 MiB
    float2* tmpA = (float2*)ws;
    float2* outb = (float2*)(ws + volB);
    float*  Wl   = (float*)(ws + 2 * volB);

    const dim3 rowgrid(NLEDS * 512 / 4);   // 12288 blocks of 256 (4 rows each)
    const dim3 colgrid(512 / 8, NLEDS);    // 64 col-tiles x 96 LEDs

    // P1: tmpA = rowFFT(pw * field)
    fpm_rowfft_kernel<<<rowgrid, 256, 0, stream>>>(pw, x, nullptr, nullptr, tmpA, 0);
    // P2: outb = col-iFFT( P * colFFT(tmpA) )
    fpm_colpass_kernel<<<colgrid, 256, 0, stream>>>(tmpA, P, nullptr, outb, 0);
    // P3: outb = row-iFFT(outb)   ("out" per LED)
    fpm_rowfft_kernel<<<rowgrid, 256, 0, stream>>>(nullptr, nullptr, outb, nullptr, outb, 1);
    // P4: Wl = C^T @ Ajx(C@|out|^2, |C@meas|)   -- WMMA f32 16x16x4
    fpm_wmma_einsum_kernel<<<HWPIX / 128, 256, 0, stream>>>(outb, meas, C, Wl);
    // P5: tmpA = rowFFT(conj(pw) * out * Wl)
    fpm_rowfft_kernel<<<rowgrid, 256, 0, stream>>>(pw, nullptr, outb, Wl, tmpA, 2);
    // P6: tmpA = col-iFFT( pupils * colFFT(tmpA) )
    fpm_colpass_kernel<<<colgrid, 256, 0, stream>>>(tmpA, nullptr, pupils, tmpA, 1);
    // P7: d_out = field - ALPHA * sum_l row-iFFT(tmpA)
    fpm_rowifft_accum_kernel<<<512 / 4, 256, 0, stream>>>(tmpA, x, (float2*)d_out);
}